// MinkowskiGlobalAveragePooling_19155554140413
// MI455X (gfx1250) — compile-verified
//
#include <hip/hip_runtime.h>

typedef __attribute__((ext_vector_type(2))) float v2f;
typedef __attribute__((ext_vector_type(8))) float v8f;

#define FEAT 64
#define NSEG 32
#define WG_THREADS 256
#define WAVES_PER_WG 8
#define SLAB (NSEG * FEAT)   // 2048 floats per workgroup partial

__global__ void zero_ovf_kernel(float* __restrict__ ovf) {
  int i = blockIdx.x * blockDim.x + threadIdx.x;
  if (i < SLAB) ovf[i] = 0.0f;
}

__global__ __launch_bounds__(WG_THREADS)
void pool_partial_kernel(const float* __restrict__ feat,
                         const int*   __restrict__ bidx,
                         float*       __restrict__ slabs,  // [W][32][64]
                         float*       __restrict__ ovf,    // [32][64]
                         int n, int rowsPerWG)
{
  __shared__ float red[16 * FEAT];
  const int tid  = threadIdx.x;
  const int lane = tid & 31;
  const int wave = tid >> 5;
  const int m    = lane & 15;   // M-row of A / N-col of B for this lane
  const int hi   = lane >> 4;   // lane half selects K within each operand VGPR

  for (int i = tid; i < 16 * FEAT; i += WG_THREADS) red[i] = 0.0f;
  __syncthreads();

  const int wg      = blockIdx.x;
  const int wgStart = wg * rowsPerWG;                 // < n by construction
  const int wgEnd   = min(wgStart + rowsPerWG, n);
  const int segBase = bidx[wgStart];                  // window [segBase, segBase+16)

  const int rowsWG      = wgEnd - wgStart;
  const int rowsPerWave = (((rowsWG + WAVES_PER_WG - 1) / WAVES_PER_WG) + 3) & ~3;
  const int wStart      = wgStart + wave * rowsPerWave;
  const int wEnd        = min(wStart + rowsPerWave, wgEnd);

  v8f acc0 = {}, acc1 = {}, acc2 = {}, acc3 = {};

  // K mapping shared by A and B: .x -> k = hi, .y -> k = 2 + hi (permutation
  // identical on both operands, so the contraction over K is correct).
  const int kx = hi;
  const int ky = 2 + hi;

  int r = wStart;
  for (; r + 4 <= wEnd; r += 4) {
    const int ix = bidx[r + kx];
    const int iy = bidx[r + ky];
    const float* fx = feat + (size_t)(r + kx) * FEAT;
    const float* fy = feat + (size_t)(r + ky) * FEAT;

    v2f a, b0, b1, b2, b3;
    a.x = (ix == segBase + m) ? 1.0f : 0.0f;   // one-hot selector A[m][k]
    a.y = (iy == segBase + m) ? 1.0f : 0.0f;
    b0.x = __builtin_nontemporal_load(fx + m);
    b0.y = __builtin_nontemporal_load(fy + m);
    b1.x = __builtin_nontemporal_load(fx + 16 + m);
    b1.y = __builtin_nontemporal_load(fy + 16 + m);
    b2.x = __builtin_nontemporal_load(fx + 32 + m);
    b2.y = __builtin_nontemporal_load(fy + 32 + m);
    b3.x = __builtin_nontemporal_load(fx + 48 + m);
    b3.y = __builtin_nontemporal_load(fy + 48 + m);

    acc0 = __builtin_amdgcn_wmma_f32_16x16x4_f32(false, a, false, b0, (short)0, acc0, false, false);
    acc1 = __builtin_amdgcn_wmma_f32_16x16x4_f32(false, a, false, b1, (short)0, acc1, false, false);
    acc2 = __builtin_amdgcn_wmma_f32_16x16x4_f32(false, a, false, b2, (short)0, acc2, false, false);
    acc3 = __builtin_amdgcn_wmma_f32_16x16x4_f32(false, a, false, b3, (short)0, acc3, false, false);

    // Rare: row's segment outside the 16-wide window (one-hot already zeroed
    // its WMMA contribution) -> accumulate via global atomics into overflow.
    if (__ballot(((unsigned)(ix - segBase) >= 16u) || ((unsigned)(iy - segBase) >= 16u))) {
      for (int k = 0; k < 4; ++k) {
        int s = bidx[r + k];                       // uniform per wave
        if ((unsigned)(s - segBase) >= 16u) {
          const float* fr = feat + (size_t)(r + k) * FEAT;
          atomicAdd(&ovf[s * FEAT + 2 * lane + 0], fr[2 * lane + 0]);
          atomicAdd(&ovf[s * FEAT + 2 * lane + 1], fr[2 * lane + 1]);
        }
      }
    }
  }
  // Tail rows (left over when wEnd - wStart is not a multiple of 4): overflow path.
  for (; r < wEnd; ++r) {
    int s = bidx[r];
    const float* fr = feat + (size_t)r * FEAT;
    atomicAdd(&ovf[s * FEAT + 2 * lane + 0], fr[2 * lane + 0]);
    atomicAdd(&ovf[s * FEAT + 2 * lane + 1], fr[2 * lane + 1]);
  }

  // C layout: VGPR j -> M = j + 8*hi, N = m. Reduce 8 waves via LDS atomics.
  const int M = hi * 8;
  #pragma unroll
  for (int j = 0; j < 8; ++j) {
    atomicAdd(&red[(M + j) * FEAT + m +  0], acc0[j]);
    atomicAdd(&red[(M + j) * FEAT + m + 16], acc1[j]);
    atomicAdd(&red[(M + j) * FEAT + m + 32], acc2[j]);
    atomicAdd(&red[(M + j) * FEAT + m + 48], acc3[j]);
  }
  __syncthreads();

  // Scatter the 16-segment window into this workgroup's 32x64 slab.
  float* slab = slabs + (size_t)wg * SLAB;
  for (int i = tid; i < SLAB; i += WG_THREADS) {
    int mm = (i >> 6) - segBase;
    slab[i] = ((unsigned)mm < 16u) ? red[mm * FEAT + (i & 63)] : 0.0f;
  }
}

__device__ inline int lower_bound_dev(const int* __restrict__ a, int n, int v) {
  int lo = 0, hi = n;
  while (lo < hi) {
    int mid = (lo + hi) >> 1;
    if (a[mid] < v) lo = mid + 1; else hi = mid;
  }
  return lo;
}

__global__ void finalize_kernel(const float* __restrict__ slabs,
                                const float* __restrict__ ovf,
                                const int*   __restrict__ bidx,
                                float*       __restrict__ out,
                                int n, int W)
{
  int e = blockIdx.x * blockDim.x + threadIdx.x;
  if (e >= SLAB) return;
  int s = e >> 6;
  int cnt = lower_bound_dev(bidx, n, s + 1) - lower_bound_dev(bidx, n, s);
  float sum = ovf[e];
  int w = 0;
  for (; w + 4 <= W; w += 4) {
    sum += slabs[(size_t)(w + 0) * SLAB + e];
    sum += slabs[(size_t)(w + 1) * SLAB + e];
    sum += slabs[(size_t)(w + 2) * SLAB + e];
    sum += slabs[(size_t)(w + 3) * SLAB + e];
  }
  for (; w < W; ++w) sum += slabs[(size_t)w * SLAB + e];
  out[e] = sum / (float)cnt;
}

extern "C" void kernel_launch(void* const* d_in, const int* in_sizes, int n_in,
                              void* d_out, int out_size, void* d_ws, size_t ws_size,
                              hipStream_t stream)
{
  const float* feat = (const float*)d_in[0];
  const int*   bidx = (const int*)d_in[1];
  float*       out  = (float*)d_out;
  const int n = in_sizes[1];

  // Workgroup count: ~4096 rows per WG, capped by workspace capacity
  // (W slabs of 2048 floats + one overflow slab).
  long long maxW = ((long long)(ws_size / sizeof(float)) - SLAB) / SLAB;
  if (maxW < 1) maxW = 1;
  int W = (n + 4095) / 4096;
  if ((long long)W > maxW) W = (int)maxW;
  if (W < 1) W = 1;
  int rowsPerWG = (n + W - 1) / W;
  W = (n + rowsPerWG - 1) / rowsPerWG;   // every workgroup owns >=1 row

  float* slabs = (float*)d_ws;
  float* ovf   = slabs + (size_t)W * SLAB;

  zero_ovf_kernel<<<(SLAB + 255) / 256, 256, 0, stream>>>(ovf);
  pool_partial_kernel<<<W, WG_THREADS, 0, stream>>>(feat, bidx, slabs, ovf, n, rowsPerWG);
  finalize_kernel<<<(SLAB + 255) / 256, 256, 0, stream>>>(slabs, ovf, bidx, out, n, W);
}